// MoEGate_19224273617584
// MI455X (gfx1250) — compile-verified
//
#include <hip/hip_runtime.h>
#include <hip/hip_bf16.h>

typedef __attribute__((ext_vector_type(2))) float v2f;
typedef __attribute__((ext_vector_type(8))) float v8f;

// Problem constants (from reference)
constexpr int Bc = 16, Sc = 8192, Dc = 128, Ec = 64, Kc = 2;
constexpr int NTOK = Bc * Sc;            // 131072 tokens
constexpr int TPB  = 128;                // 4 waves per block
constexpr int NW   = TPB / 32;           // waves per block
constexpr int TOKW = 32;                 // tokens per wave
constexpr int NBLK = NTOK / (NW * TOKW); // 1024 blocks

// d_out layout (flat floats, reference tuple order)
constexpr size_t OFF_IDX   = 0;
constexpr size_t OFF_W     = (size_t)NTOK * Kc;           // 262144
constexpr size_t OFF_AUX   = (size_t)NTOK * Kc * 2;       // 524288
constexpr size_t OFF_PROBS = OFF_AUX + 1;                 // 524289 (only 4B aligned!)

__global__ __launch_bounds__(TPB) void moe_gate_kernel(
    const float* __restrict__ x, const float* __restrict__ W,
    const float* __restrict__ bvec, float* __restrict__ out,
    float* __restrict__ ws)
{
    __shared__ float sS[NW][TOKW][Ec + 1];  // +1 pad: kills bank conflicts on row reads
    __shared__ float sMx[NW][TOKW];
    __shared__ float sInv[NW][TOKW];
    __shared__ float sFreq[Ec];

    const int tid  = threadIdx.x;
    const int wv   = tid >> 5;
    const int lane = tid & 31;
    const int half = lane >> 4;     // 0: K=k,k+1 ; 1: K=k+2,k+3 (ISA 16x4 A layout)
    const int l16  = lane & 15;
    const int m0   = blockIdx.x * (NW * TOKW) + wv * TOKW;

    if (tid < Ec) sFreq[tid] = 0.0f;

    // ---- WMMA phase: scores[32 tok][64 exp] = x @ W^T ----
    const int koff = half * 2;
    const float* xlo = x + (size_t)(m0 + l16)      * Dc + koff;  // A frag, M-tile 0
    const float* xhi = x + (size_t)(m0 + 16 + l16) * Dc + koff;  // A frag, M-tile 1
    const float* wp0 = W + (size_t)(l16)           * Dc + koff;  // B frags (B[k][n]=W[n][k])
    const float* wp1 = W + (size_t)(16 + l16)      * Dc + koff;
    const float* wp2 = W + (size_t)(32 + l16)      * Dc + koff;
    const float* wp3 = W + (size_t)(48 + l16)      * Dc + koff;

    v8f acc[2][4];
    #pragma unroll
    for (int mt = 0; mt < 2; ++mt)
        #pragma unroll
        for (int nt = 0; nt < 4; ++nt)
            acc[mt][nt] = (v8f)(0.0f);

    #pragma unroll 8
    for (int kk = 0; kk < Dc / 4; ++kk) {
        const int o = kk * 4;
        v2f a0 = *(const v2f*)(xlo + o);
        v2f a1 = *(const v2f*)(xhi + o);
        v2f b0 = *(const v2f*)(wp0 + o);
        v2f b1 = *(const v2f*)(wp1 + o);
        v2f b2 = *(const v2f*)(wp2 + o);
        v2f b3 = *(const v2f*)(wp3 + o);
        acc[0][0] = __builtin_amdgcn_wmma_f32_16x16x4_f32(false, a0, false, b0, (short)0, acc[0][0], false, false);
        acc[0][1] = __builtin_amdgcn_wmma_f32_16x16x4_f32(false, a0, false, b1, (short)0, acc[0][1], false, false);
        acc[0][2] = __builtin_amdgcn_wmma_f32_16x16x4_f32(false, a0, false, b2, (short)0, acc[0][2], false, false);
        acc[0][3] = __builtin_amdgcn_wmma_f32_16x16x4_f32(false, a0, false, b3, (short)0, acc[0][3], false, false);
        acc[1][0] = __builtin_amdgcn_wmma_f32_16x16x4_f32(false, a1, false, b0, (short)0, acc[1][0], false, false);
        acc[1][1] = __builtin_amdgcn_wmma_f32_16x16x4_f32(false, a1, false, b1, (short)0, acc[1][1], false, false);
        acc[1][2] = __builtin_amdgcn_wmma_f32_16x16x4_f32(false, a1, false, b2, (short)0, acc[1][2], false, false);
        acc[1][3] = __builtin_amdgcn_wmma_f32_16x16x4_f32(false, a1, false, b3, (short)0, acc[1][3], false, false);
    }

    // ---- Scatter scores (+bias) to LDS. D layout: vgpr r -> M=r (lanes 0-15) / M=r+8 (16-31), N=l16+16*nt
    float bv[4];
    #pragma unroll
    for (int nt = 0; nt < 4; ++nt) bv[nt] = bvec[nt * 16 + l16];
    #pragma unroll
    for (int mt = 0; mt < 2; ++mt)
        #pragma unroll
        for (int nt = 0; nt < 4; ++nt)
            #pragma unroll
            for (int r = 0; r < 8; ++r)
                sS[wv][16 * mt + r + 8 * half][nt * 16 + l16] = acc[mt][nt][r] + bv[nt];

    __syncthreads();

    // ---- Per-token phase: each lane owns one token ----
    {
        const int t = lane;
        const float* row = &sS[wv][t][0];
        float mx = -3.402823466e38f;
        #pragma unroll 8
        for (int e = 0; e < Ec; ++e) mx = fmaxf(mx, row[e]);

        float sum = 0.0f;
        float s1 = -3.402823466e38f, s2 = -3.402823466e38f;
        int i1 = 0, i2 = 0;
        #pragma unroll 4
        for (int e = 0; e < Ec; ++e) {
            float v = row[e];
            sum += __expf(v - mx);
            if (v > s1)      { s2 = s1; i2 = i1; s1 = v; i1 = e; }
            else if (v > s2) { s2 = v; i2 = e; }
        }
        const float inv = 1.0f / sum;
        const float w1 = __expf(s1 - mx) * inv;
        const float w2 = __expf(s2 - mx) * inv;

        const size_t token = (size_t)m0 + t;
        out[OFF_IDX + token * 2 + 0] = (float)i1;
        out[OFF_IDX + token * 2 + 1] = (float)i2;
        out[OFF_W   + token * 2 + 0] = w1;
        out[OFF_W   + token * 2 + 1] = w2;

        sMx[wv][t]  = mx;
        sInv[wv][t] = inv;
        atomicAdd(&sFreq[i1], w1);
        atomicAdd(&sFreq[i2], w2);
    }

    __syncthreads();

    // ---- Coalesced probs store (b32: probs region is only 4B-aligned) ----
    {
        float* probs = out + OFF_PROBS + (size_t)m0 * Ec;
        #pragma unroll 8
        for (int i = 0; i < (TOKW * Ec) / 32; ++i) {
            const int f  = i * 32 + lane;
            const int tl = f >> 6;
            const int e  = f & (Ec - 1);
            probs[f] = __expf(sS[wv][tl][e] - sMx[wv][tl]) * sInv[wv][tl];
        }
    }

    // ---- Per-block expert-frequency partials (deterministic cross-block reduce) ----
    if (tid < Ec) ws[(size_t)blockIdx.x * Ec + tid] = sFreq[tid];
}

__global__ __launch_bounds__(64) void moe_aux_kernel(const float* __restrict__ ws,
                                                     float* __restrict__ out)
{
    __shared__ float sq[Ec];
    const int e = threadIdx.x;  // 64 threads
    float s = 0.0f;
    for (int bk = 0; bk < NBLK; ++bk) s += ws[(size_t)bk * Ec + e];
    const float freq = s * (1.0f / (float)NTOK);
    const float d = freq - (1.0f / (float)Ec);
    sq[e] = d * d;
    __syncthreads();
    if (e == 0) {
        float t = 0.0f;
        for (int i = 0; i < Ec; ++i) t += sq[i];
        out[OFF_AUX] = t * (0.01f / (float)Ec);  // alpha * mean
    }
}

extern "C" void kernel_launch(void* const* d_in, const int* in_sizes, int n_in,
                              void* d_out, int out_size, void* d_ws, size_t ws_size,
                              hipStream_t stream) {
    const float* x = (const float*)d_in[0];   // [B,S,D]
    const float* W = (const float*)d_in[1];   // [E,D]
    const float* b = (const float*)d_in[2];   // [E]
    float* out = (float*)d_out;
    float* ws  = (float*)d_ws;                // NBLK*Ec floats of expert partial sums

    moe_gate_kernel<<<NBLK, TPB, 0, stream>>>(x, W, b, out, ws);
    moe_aux_kernel<<<1, 64, 0, stream>>>(ws, out);
}